// VectorQuantizer_1022202216471
// MI455X (gfx1250) — compile-verified
//
#include <hip/hip_runtime.h>
#include <hip/hip_bf16.h>
#include <math.h>

typedef __attribute__((ext_vector_type(2))) float v2f;
typedef __attribute__((ext_vector_type(8))) float v8f;

#define CC   64
#define HWsz 1024
#define NN   65536
#define KK   512

// ---------------- kernel: ||w||^2 per codebook row + zero histogram ----------------
__global__ __launch_bounds__(512) void k_wnorm_init(const float* __restrict__ w,
                                                    float* __restrict__ wnorm,
                                                    unsigned int* __restrict__ counts) {
    int k = threadIdx.x;            // 512 threads, one block
    float s = 0.f;
    #pragma unroll
    for (int c = 0; c < CC; ++c) { float v = w[k * CC + c]; s += v * v; }
    wnorm[k]  = s;
    counts[k] = 0u;
}

// ---------------- kernel: ||x||^2 per flattened NHWC row ----------------
__global__ __launch_bounds__(256) void k_xnorm(const float* __restrict__ in,
                                               float* __restrict__ xnorm) {
    int n  = blockIdx.x * 256 + threadIdx.x;   // 65536 threads
    int b  = n >> 10;
    int hw = n & (HWsz - 1);
    const float* p = in + (size_t)b * CC * HWsz + hw;
    float s = 0.f;
    #pragma unroll
    for (int c = 0; c < CC; ++c) { float v = p[c * HWsz]; s += v * v; }  // coalesced across threads
    xnorm[n] = s;
}

// ---------------- kernel: distances via V_WMMA_F32_16X16X4_F32 ----------------
// block: 256 threads (8 waves). tile: 128 rows (n) x 64 cols (k), full C=64 depth.
__global__ __launch_bounds__(256) void k_dist_wmma(const float* __restrict__ in,
                                                   const float* __restrict__ w,
                                                   const float* __restrict__ xnorm,
                                                   const float* __restrict__ wnorm,
                                                   float* __restrict__ dist) {
    __shared__ float lds_a[128][65];   // [n][c], pad 65 -> conflict-free column reads
    __shared__ float lds_b[64][65];    // [c][k], pad 65 -> conflict-free transposed writes
    __shared__ float lds_xn[128];

    const int t  = threadIdx.x;
    const int n0 = blockIdx.x * 128;   // 512 n-tiles
    const int k0 = blockIdx.y * 64;    // 8 k-tiles
    const int b   = n0 >> 10;          // 128 | 1024 -> tile never crosses batch
    const int hw0 = n0 & (HWsz - 1);

    // A tile: NCHW -> [n][c]; reads coalesced along hw, LDS transpose
    const float* ain = in + (size_t)b * CC * HWsz + hw0;
    #pragma unroll
    for (int i = 0; i < 32; ++i) {
        int idx = t + 256 * i;         // 8192 elements
        int c  = idx >> 7;
        int hw = idx & 127;
        lds_a[hw][c] = ain[c * HWsz + hw];
    }
    // B tile: weight[k0+kc][c] -> lds_b[c][kc]; reads coalesced along c
    #pragma unroll
    for (int i = 0; i < 16; ++i) {
        int idx = t + 256 * i;         // 4096 elements
        int kc = idx >> 6;
        int c  = idx & 63;
        lds_b[c][kc] = w[(size_t)(k0 + kc) * CC + c];
    }
    if (t < 128) lds_xn[t] = xnorm[n0 + t];
    __syncthreads();

    const int wave = t >> 5;
    const int lane = t & 31;
    const int lsub = lane & 15;
    const int hi   = lane >> 4;             // 0: K=0,1 / rows M   1: K=2,3 / rows M+8
    const int row  = wave * 16 + lsub;      // A fragment row (M = lane%16)

    v8f acc[4] = {v8f{}, v8f{}, v8f{}, v8f{}};

    #pragma unroll
    for (int cs = 0; cs < CC; cs += 4) {
        // A 16x4 fragment: lane<16 holds K=cs+0,cs+1 ; lane>=16 holds K=cs+2,cs+3
        v2f a;
        a.x = lds_a[row][cs + 2 * hi];
        a.y = lds_a[row][cs + 2 * hi + 1];
        #pragma unroll
        for (int ks = 0; ks < 4; ++ks) {
            // B 4x16 fragment: VGPR0 rows K=0(/2), VGPR1 rows K=1(/3); N = lane%16
            v2f bb;
            bb.x = lds_b[cs + 2 * hi][ks * 16 + lsub];
            bb.y = lds_b[cs + 2 * hi + 1][ks * 16 + lsub];
            acc[ks] = __builtin_amdgcn_wmma_f32_16x16x4_f32(
                false, a, false, bb, (short)0, acc[ks], false, false);
        }
    }

    // distances = ||x||^2 + ||w||^2 - 2 * (x . w)
    #pragma unroll
    for (int ks = 0; ks < 4; ++ks) {
        int k = k0 + ks * 16 + lsub;        // column fixed per lane
        float wn = wnorm[k];
        #pragma unroll
        for (int i = 0; i < 8; ++i) {       // C/D layout: VGPR i -> rows i (lanes 0-15) / i+8 (lanes 16-31)
            int ml = wave * 16 + hi * 8 + i;
            float d = lds_xn[ml] + wn - 2.0f * acc[ks][i];
            dist[(size_t)(n0 + ml) * KK + k] = d;
        }
    }
}

// ---------------- kernel: argmin over K=512 (one wave32 per row) + one-hot ----------------
__global__ __launch_bounds__(256) void k_argmin(const float* __restrict__ dist,
                                                float* __restrict__ enc,
                                                float* __restrict__ eidx,
                                                int* __restrict__ idxws,
                                                unsigned int* __restrict__ counts) {
    int wave = threadIdx.x >> 5;
    int lane = threadIdx.x & 31;
    int n    = blockIdx.x * 8 + wave;       // 8192 blocks
    const float* dr = dist + (size_t)n * KK;

    float bv = 3.402823466e38f;
    int   bi = 0;
    #pragma unroll
    for (int j = 0; j < 16; ++j) {
        int k = lane + j * 32;              // coalesced
        float v = dr[k];
        if (v < bv) { bv = v; bi = k; }     // per-lane k strictly increasing -> first-min kept
    }
    #pragma unroll
    for (int off = 16; off; off >>= 1) {    // wave32 xor reduction; tie -> smaller index (jnp.argmin)
        float ov = __shfl_xor(bv, off, 32);
        int   oi = __shfl_xor(bi, off, 32);
        if (ov < bv || (ov == bv && oi < bi)) { bv = ov; bi = oi; }
    }
    if (lane == 0) {
        idxws[n] = bi;
        eidx[n]  = (float)bi;
        atomicAdd(&counts[bi], 1u);
    }
    float* er = enc + (size_t)n * KK;
    #pragma unroll
    for (int j = 0; j < 16; ++j) {
        int k = lane + j * 32;
        er[k] = (k == bi) ? 1.0f : 0.0f;    // coalesced, overwrites full row
    }
}

// ---------------- kernel: gather codebook rows into NCHW output ----------------
__global__ __launch_bounds__(256) void k_quant(const float* __restrict__ w,
                                               const int* __restrict__ idxws,
                                               float* __restrict__ q) {
    int tid = blockIdx.x * 256 + threadIdx.x;   // 4,194,304 threads
    int hw  = tid & (HWsz - 1);
    int c   = (tid >> 10) & 63;
    int b   = tid >> 16;
    int n   = (b << 10) | hw;
    q[tid]  = w[idxws[n] * CC + c];             // idx loads coalesced, codebook (128KB) L2-resident
}

// ---------------- kernel: per-batch loss = 1.25 * mean((q - x)^2) ----------------
__global__ __launch_bounds__(256) void k_loss(const float* __restrict__ in,
                                              const float* __restrict__ q,
                                              float* __restrict__ loss) {
    __shared__ float red[256];
    int b = blockIdx.x;                          // 64 blocks
    const float* ib = in + (size_t)b * CC * HWsz;
    const float* qb = q  + (size_t)b * CC * HWsz;
    float s = 0.f;
    for (int j = threadIdx.x; j < CC * HWsz; j += 256) {
        float d = qb[j] - ib[j];
        s += d * d;
    }
    red[threadIdx.x] = s;
    __syncthreads();
    for (int o = 128; o; o >>= 1) {
        if (threadIdx.x < o) red[threadIdx.x] += red[threadIdx.x + o];
        __syncthreads();
    }
    if (threadIdx.x == 0) loss[b] = 1.25f * red[0] * (1.0f / (CC * HWsz));
}

// ---------------- kernel: perplexity from histogram ----------------
__global__ __launch_bounds__(512) void k_perp(const unsigned int* __restrict__ counts,
                                              float* __restrict__ perp) {
    __shared__ float red[512];
    int t = threadIdx.x;
    float p = (float)counts[t] * (1.0f / (float)NN);
    red[t] = p * logf(p + 1e-10f);
    __syncthreads();
    for (int o = 256; o; o >>= 1) {
        if (t < o) red[t] += red[t + o];
        __syncthreads();
    }
    if (t == 0) *perp = expf(-red[0]);
}

extern "C" void kernel_launch(void* const* d_in, const int* in_sizes, int n_in,
                              void* d_out, int out_size, void* d_ws, size_t ws_size,
                              hipStream_t stream) {
    const float* in = (const float*)d_in[0];   // [64,64,32,32]
    const float* w  = (const float*)d_in[1];   // [512,64]
    float* out = (float*)d_out;

    // output tuple layout (flat, return order)
    float* q    = out;                          // 4,194,304
    float* loss = out + 4194304;                // 64
    float* perp = out + 4194368;                // 1
    float* enc  = out + 4194369;                // 33,554,432
    float* eidx = out + 37748801;               // 65,536
    float* dist = out + 37814337;               // 33,554,432

    // workspace
    float*        xnorm  = (float*)d_ws;                 // [65536]
    float*        wnorm  = xnorm + NN;                   // [512]
    unsigned int* counts = (unsigned int*)(wnorm + KK);  // [512]
    int*          idxws  = (int*)(counts + KK);          // [65536]

    k_wnorm_init<<<1, 512, 0, stream>>>(w, wnorm, counts);
    k_xnorm<<<NN / 256, 256, 0, stream>>>(in, xnorm);
    k_dist_wmma<<<dim3(NN / 128, KK / 64), 256, 0, stream>>>(in, w, xnorm, wnorm, dist);
    k_argmin<<<NN / 8, 256, 0, stream>>>(dist, enc, eidx, idxws, counts);
    k_quant<<<(64 * CC * HWsz) / 256, 256, 0, stream>>>(w, idxws, q);
    k_loss<<<64, 256, 0, stream>>>(in, q, loss);
    k_perp<<<1, 512, 0, stream>>>(counts, perp);
}